// Model_85066122265072
// MI455X (gfx1250) — compile-verified
//
#include <hip/hip_runtime.h>
#include <hip/hip_bf16.h>

typedef __attribute__((ext_vector_type(2))) float v2f;
typedef __attribute__((ext_vector_type(8))) float v8f;

#define Bb 8
#define Rr 128
#define Ss 512
#define Dd 33
#define Ee 128
#define Hh 4
#define EKk 32

// ---------------------------------------------------------------------------
// sa = sigmoid(stride @ Wr.T + br)   (33 outputs; also written to d_out tail)
// ---------------------------------------------------------------------------
__global__ void sa_kernel(const float* __restrict__ stridev,
                          const float* __restrict__ Wr,
                          const float* __restrict__ br,
                          float* __restrict__ sa_ws,
                          float* __restrict__ out_tail) {
  int d = threadIdx.x;
  if (d < Dd) {
    float z = br[d];
    for (int j = 0; j < Dd; ++j) z += stridev[j] * Wr[d * Dd + j];
    float s = 1.0f / (1.0f + __expf(-z));
    sa_ws[d] = s;
    out_tail[d] = s;
  }
}

// ---------------------------------------------------------------------------
// Generic C = A(MxK) * W(NxK)^T + bias, f32 WMMA 16x16x4, one wave per tile.
// N = Ntiles*16.  Used for Q-proj (K=128), K-proj (K=128), out-proj (K=132).
// ---------------------------------------------------------------------------
__global__ __launch_bounds__(256) void gemm_wt_kernel(
    const float* __restrict__ A, const float* __restrict__ W,
    const float* __restrict__ bias, float* __restrict__ C,
    int M, int K, int Ntiles) {
  int wave = threadIdx.x >> 5;
  int lane = threadIdx.x & 31;
  int tile = blockIdx.x * 8 + wave;
  int mt = tile / Ntiles;
  int nt = tile % Ntiles;
  if (mt * 16 >= M) return;  // wave-uniform: EXEC stays all-ones for WMMA

  int rc = lane & 15;   // A-row / W-row (B-col) index
  int hi = lane >> 4;   // K sub-pair select (K offset +2 for upper half)

  const float* Abase = A + (size_t)(mt * 16 + rc) * K + hi * 2;
  const float* Wbase = W + (size_t)(nt * 16 + rc) * K + hi * 2;

  v8f c = {};
  for (int k0 = 0; k0 < K; k0 += 4) {
    v2f a = *(const v2f*)(Abase + k0);  // A[row, k0+hi*2 .. +1]
    v2f b = *(const v2f*)(Wbase + k0);  // B[k0+hi*2 .. +1, col] = W[col, k]
    c = __builtin_amdgcn_wmma_f32_16x16x4_f32(false, a, false, b, (short)0, c,
                                              false, false);
  }
  int N = Ntiles * 16;
  float bv = bias[nt * 16 + rc];
  float* Cb = C + (size_t)(mt * 16 + hi * 8) * N + nt * 16 + rc;
  for (int i = 0; i < 8; ++i) Cb[(size_t)i * N] = c[i] + bv;
}

// ---------------------------------------------------------------------------
// scores[b,h,r,s] = (q[b,r,h*32:] . k[b,s,h*32:]) / sqrt(32)
// One wave per 16x16 (r,s) tile; K=32 -> 8 WMMA steps.
// ---------------------------------------------------------------------------
__global__ __launch_bounds__(256) void scores_kernel(
    const float* __restrict__ qb,   // (B,R,128)
    const float* __restrict__ kb,   // (B,S,128)
    float* __restrict__ scores) {   // (B,H,R,S), pre-scaled
  int wave = threadIdx.x >> 5;
  int lane = threadIdx.x & 31;
  int tile = blockIdx.x * 8 + wave;  // 8192 tiles total
  int st = tile & 31;
  int rt = (tile >> 5) & 7;
  int h  = (tile >> 8) & 3;
  int b  = tile >> 10;

  int rc = lane & 15;
  int hi = lane >> 4;

  const float* Abase = qb + ((size_t)b * Rr + rt * 16 + rc) * Ee + h * EKk + hi * 2;
  const float* Bbase = kb + ((size_t)b * Ss + st * 16 + rc) * Ee + h * EKk + hi * 2;

  v8f c = {};
  for (int k0 = 0; k0 < EKk; k0 += 4) {
    v2f a = *(const v2f*)(Abase + k0);
    v2f bfr = *(const v2f*)(Bbase + k0);
    c = __builtin_amdgcn_wmma_f32_16x16x4_f32(false, a, false, bfr, (short)0, c,
                                              false, false);
  }
  const float scale = 0.17677669529663687f;  // 1/sqrt(32)
  float* out = scores + (((size_t)(b * Hh + h) * Rr) + rt * 16) * Ss + st * 16;
  for (int i = 0; i < 8; ++i)
    out[(size_t)(i + hi * 8) * Ss + rc] = c[i] * scale;
}

// ---------------------------------------------------------------------------
// Per (b,r): masked softmax over s for every (h,d), weighted sum with value.
// x[b,r,h*33+d] = sum_s e[h,s]*a[s,d]*v[s,d] / sum_s e[h,s]*a[s,d]
// a[s,d] = mask[s,d] * (|tt[s]-qt[r]| <= sa[d]); all-masked -> mean_s v[s,d].
// LDS: value 66KB + mask bytes 16.5KB + exp 8KB + dt 2KB (< 320KB/WGP).
// ---------------------------------------------------------------------------
__global__ __launch_bounds__(256) void attn_kernel(
    const float* __restrict__ scores,  // (B,H,R,S)
    const float* __restrict__ value,   // (B,S,D)
    const float* __restrict__ mask,    // (B,S,D)
    const float* __restrict__ qt,      // (R)
    const float* __restrict__ tt,      // (B,S)
    const float* __restrict__ sa,      // (D)
    float* __restrict__ xbuf) {        // (B,R,H*D)
  __shared__ float v_sh[Ss * Dd];
  __shared__ unsigned char m_sh[Ss * Dd];
  __shared__ float e_sh[Hh][Ss];
  __shared__ float dt_sh[Ss];
  __shared__ float red[Hh][64];

  int t = threadIdx.x;
  int b = blockIdx.x >> 7;
  int r = blockIdx.x & 127;

  // Stage value / mask for this batch (fully coalesced flat copy).
  const float* vb = value + (size_t)b * Ss * Dd;
  const float* mb = mask + (size_t)b * Ss * Dd;
  for (int i = t; i < Ss * Dd; i += 256) {
    v_sh[i] = vb[i];
    m_sh[i] = (mb[i] != 0.0f) ? 1 : 0;
  }
  float qtr = qt[r];
  for (int s = t; s < Ss; s += 256) dt_sh[s] = fabsf(tt[b * Ss + s] - qtr);

  // Row max + exp per head: 64 threads per head, 8 scores each.
  int h = t >> 6;
  int l64 = t & 63;
  const float* scb = scores + ((size_t)(b * Hh + h) * Rr + r) * Ss;
  float sc_loc[8];
  float lmax = -3.0e38f;
  for (int i = 0; i < 8; ++i) {
    float v = scb[l64 + 64 * i];
    sc_loc[i] = v;
    lmax = fmaxf(lmax, v);
  }
  red[h][l64] = lmax;
  __syncthreads();
  for (int off = 32; off > 0; off >>= 1) {
    if (l64 < off) red[h][l64] = fmaxf(red[h][l64], red[h][l64 + off]);
    __syncthreads();
  }
  float mh = red[h][0];
  for (int i = 0; i < 8; ++i)
    e_sh[h][l64 + 64 * i] = __expf(sc_loc[i] - mh);
  __syncthreads();

  // Accumulate: one thread per (h,d) pair (132 active).
  if (t < Hh * Dd) {
    int ha = t / Dd;
    int d = t % Dd;
    float sad = sa[d];
    const float* eh = e_sh[ha];
    float num = 0.f, den = 0.f, vsum = 0.f;
    for (int s = 0; s < Ss; ++s) {
      float v = v_sh[s * Dd + d];
      vsum += v;
      if (dt_sh[s] <= sad && m_sh[s * Dd + d]) {
        float w = eh[s];
        num += w * v;
        den += w;
      }
    }
    float x = (den > 0.f) ? (num / den) : (vsum * (1.0f / Ss));
    xbuf[((size_t)b * Rr + r) * (Hh * Dd) + ha * Dd + d] = x;
  }
}

// ---------------------------------------------------------------------------
extern "C" void kernel_launch(void* const* d_in, const int* in_sizes, int n_in,
                              void* d_out, int out_size, void* d_ws,
                              size_t ws_size, hipStream_t stream) {
  const float* query = (const float*)d_in[0];
  const float* key   = (const float*)d_in[1];
  const float* value = (const float*)d_in[2];
  const float* mask  = (const float*)d_in[3];
  const float* qt    = (const float*)d_in[4];
  const float* tt    = (const float*)d_in[5];
  const float* strv  = (const float*)d_in[6];
  const float* Wq    = (const float*)d_in[7];
  const float* bq    = (const float*)d_in[8];
  const float* Wk    = (const float*)d_in[9];
  const float* bk    = (const float*)d_in[10];
  const float* Wr    = (const float*)d_in[11];
  const float* br    = (const float*)d_in[12];
  const float* Wo    = (const float*)d_in[13];
  const float* bo    = (const float*)d_in[14];
  float* out = (float*)d_out;

  // Workspace layout (floats): sa(64) | qbuf | kbuf | xbuf | scores  (~11.6MB)
  float* ws = (float*)d_ws;
  float* sa_ws  = ws;
  float* qbuf   = ws + 64;                 // 8*128*128   = 131072
  float* kbuf   = qbuf + 131072;           // 8*512*128   = 524288
  float* xbuf   = kbuf + 524288;           // 8*128*132   = 135168
  float* scoreb = xbuf + 135168;           // 8*4*128*512 = 2097152

  sa_kernel<<<1, 64, 0, stream>>>(strv, Wr, br, sa_ws, out + Bb * Rr * 128);
  // Q-proj: M=1024, K=128, N=128 -> 512 tiles / 8 waves = 64 blocks
  gemm_wt_kernel<<<64, 256, 0, stream>>>(query, Wq, bq, qbuf, Bb * Rr, Ee, 8);
  // K-proj: M=4096 -> 2048 tiles = 256 blocks
  gemm_wt_kernel<<<256, 256, 0, stream>>>(key, Wk, bk, kbuf, Bb * Ss, Ee, 8);
  // scores: 8192 tiles = 1024 blocks
  scores_kernel<<<1024, 256, 0, stream>>>(qbuf, kbuf, scoreb);
  // attention core: one block per (b,r)
  attn_kernel<<<Bb * Rr, 256, 0, stream>>>(scoreb, value, mask, qt, tt, sa_ws,
                                           xbuf);
  // out-proj: M=1024, K=132, N=128 -> 64 blocks; writes d_out directly
  gemm_wt_kernel<<<64, 256, 0, stream>>>(xbuf, Wo, bo, out, Bb * Rr, 132, 8);
}